// Time_attention_19172734009844
// MI455X (gfx1250) — compile-verified
//
#include <hip/hip_runtime.h>
#include <hip/hip_bf16.h>

// Problem constants from the reference.
#define NB 32
#define CH 64
#define TT 2048
#define ST 2052   // padded LDS row stride in floats: 2052 % 64 == 4 -> conflict-free
                  // for both row-major stores and transposed (strided) reads.

typedef float v2f __attribute__((ext_vector_type(2)));
typedef float v8f __attribute__((ext_vector_type(8)));

// ---------------------------------------------------------------------------
// Kernel 1: per-(b,t) channel sums  s[b,t] = sum_c x[b,c,t]
// ---------------------------------------------------------------------------
__global__ __launch_bounds__(256) void ta_channel_sums(const float* __restrict__ x,
                                                       float* __restrict__ s) {
  int idx = blockIdx.x * blockDim.x + threadIdx.x;   // b*TT + t
  if (idx >= NB * TT) return;
  int b = idx / TT;
  int t = idx - b * TT;
  const float* xb = x + (size_t)b * CH * TT + t;
  float acc = 0.0f;
#pragma unroll 8
  for (int c = 0; c < CH; ++c) acc += xb[(size_t)c * TT];
  s[idx] = acc;
}

// ---------------------------------------------------------------------------
// Kernel 2: fused energy -> minmax-normalize -> softmax -> attended -> out
// One workgroup = (batch b, 16-row i-block). 128 threads = 4 wave32.
// ---------------------------------------------------------------------------
__global__ __launch_bounds__(128) void ta_fused_attention(
    const float* __restrict__ x,
    const float* __restrict__ w1, const float* __restrict__ b1,
    const float* __restrict__ w2, const float* __restrict__ b2,
    const float* __restrict__ gammap,
    const float* __restrict__ s,
    float* __restrict__ out) {

  __shared__ float strip[16 * ST];   // 16 x 2048 energy/attention strip (padded)

  const int ib   = blockIdx.x;            // i-block: 0..127
  const int b    = blockIdx.y;            // batch:   0..31
  const int i0   = ib * 16;
  const int tid  = threadIdx.x;
  const int wave = tid >> 5;              // 0..3
  const int lane = tid & 31;
  const int half = lane >> 4;             // 0 for lanes 0-15, 1 for lanes 16-31
  const int l16  = lane & 15;

  // Scalar affine coefficients (8-wide dot products; cached scalar loads).
  float alpha = 0.f, beta = 0.f, delta = 0.f, eps = 0.f;
#pragma unroll
  for (int m = 0; m < 8; ++m) {
    alpha += w1[m] * w2[m];
    beta  += w1[m] * b2[m];
    delta += b1[m] * w2[m];
    eps   += b1[m] * b2[m];
  }
  const float konst = (float)CH * eps;
  const float gamma = gammap[0];

  const float* xb = x + (size_t)b * CH * TT;
  const float* sb = s + (size_t)b * TT;

  // ---------------- Phase 1: energy strip via V_WMMA_F32_16X16X4_F32 --------
  // A (16x64, M=i rows, K=channels) kept in registers.
  // f32 16x4 A layout: lanes 0-15 -> M=lane, K = kbase+{0,1}; lanes 16-31 ->
  // M=lane-16, K = kbase+{2,3}.
  v2f A[16];
#pragma unroll
  for (int ks = 0; ks < 16; ++ks) {
    int kc = ks * 4 + half * 2;                      // channel index
    A[ks].x = xb[(size_t)(kc + 0) * TT + i0 + l16];
    A[ks].y = xb[(size_t)(kc + 1) * TT + i0 + l16];
  }
  // s_i for the 8 C/D rows this lane owns (M = r + half*8).
  float si[8];
#pragma unroll
  for (int r = 0; r < 8; ++r) si[r] = sb[i0 + r + half * 8];

  for (int jt = wave; jt < TT / 16; jt += 4) {       // uniform per wave
    const int j0 = jt * 16;
    v8f acc = {};
#pragma unroll
    for (int ks = 0; ks < 16; ++ks) {
      int kc = ks * 4 + half * 2;
      v2f Bv;
      Bv.x = xb[(size_t)(kc + 0) * TT + j0 + l16];
      Bv.y = xb[(size_t)(kc + 1) * TT + j0 + l16];
      acc = __builtin_amdgcn_wmma_f32_16x16x4_f32(
          /*neg_a=*/false, A[ks], /*neg_b=*/false, Bv,
          /*c_mod=*/(short)0, acc, /*reuse_a=*/false, /*reuse_b=*/false);
    }
    const float sj = sb[j0 + l16];
#pragma unroll
    for (int r = 0; r < 8; ++r) {
      float e = alpha * acc[r] + beta * si[r] + delta * sj + konst;
      strip[(r + half * 8) * ST + j0 + l16] = e;
    }
  }
  __syncthreads();

  // ---------------- Phase 2: per-row minmax-normalize + softmax -------------
  for (int r = wave; r < 16; r += 4) {               // one wave per row
    float* row = strip + r * ST;
    float mn = __builtin_huge_valf();
    float mx = -__builtin_huge_valf();
    for (int j = lane; j < TT; j += 32) {
      float v = row[j];
      mn = fminf(mn, v);
      mx = fmaxf(mx, v);
    }
#pragma unroll
    for (int o = 16; o >= 1; o >>= 1) {
      mn = fminf(mn, __shfl_xor(mn, o, 32));
      mx = fmaxf(mx, __shfl_xor(mx, o, 32));
    }
    const float scale = 1.0f / (mx - mn + 1e-8f);
    float sum = 0.0f;
    for (int j = lane; j < TT; j += 32) {
      float w = __expf((row[j] - mn) * scale);       // softmax shift cancels
      row[j] = w;
      sum += w;
    }
#pragma unroll
    for (int o = 16; o >= 1; o >>= 1) sum += __shfl_xor(sum, o, 32);
    const float inv = 1.0f / sum;
    for (int j = lane; j < TT; j += 32) row[j] *= inv;
  }
  __syncthreads();

  // ---------------- Phase 3: attended[c,i] = sum_t X[c,t] * attn[i,t] -------
  // One wave per 16-channel c-tile. D (16x16): M = c, N = i, K = t.
  // A[m,k] = xb[(c0+m)*TT + k];  B[k,n] = attn[i0+n][k] = strip[n*ST + k].
  {
    const int c0 = wave * 16;
    v8f acc = {};
    for (int ks = 0; ks < TT / 4; ++ks) {
      const int kb = ks * 4 + half * 2;              // t index
      v2f Av, Bv;
      Av.x = xb[(size_t)(c0 + l16) * TT + kb + 0];
      Av.y = xb[(size_t)(c0 + l16) * TT + kb + 1];
      Bv.x = strip[l16 * ST + kb + 0];
      Bv.y = strip[l16 * ST + kb + 1];
      acc = __builtin_amdgcn_wmma_f32_16x16x4_f32(
          false, Av, false, Bv, (short)0, acc, false, false);
    }
    // Epilogue: out = gamma * attended + x   (layout (B, CH, T, 1) flat)
#pragma unroll
    for (int r = 0; r < 8; ++r) {
      const int c = c0 + r + half * 8;
      const size_t o = (size_t)b * CH * TT + (size_t)c * TT + i0 + l16;
      out[o] = gamma * acc[r] + x[o];
    }
  }
}

// ---------------------------------------------------------------------------
// Host launcher
// ---------------------------------------------------------------------------
extern "C" void kernel_launch(void* const* d_in, const int* in_sizes, int n_in,
                              void* d_out, int out_size, void* d_ws, size_t ws_size,
                              hipStream_t stream) {
  const float* x  = (const float*)d_in[0];
  const float* w1 = (const float*)d_in[1];
  const float* b1 = (const float*)d_in[2];
  const float* w2 = (const float*)d_in[3];
  const float* b2 = (const float*)d_in[4];
  const float* g  = (const float*)d_in[5];
  float* out = (float*)d_out;
  float* s   = (float*)d_ws;                 // NB*TT floats = 256 KB scratch

  ta_channel_sums<<<(NB * TT + 255) / 256, 256, 0, stream>>>(x, s);

  dim3 grid(TT / 16, NB);                    // (128, 32) workgroups
  ta_fused_attention<<<grid, 128, 0, stream>>>(x, w1, b1, w2, b2, g, s, out);
}